// Oracle_Multichannel_Wiener_Filtering_68135361183977
// MI455X (gfx1250) — compile-verified
//
#include <hip/hip_runtime.h>
#include <math.h>

// ---------------------------------------------------------------------------
// Oracle multichannel Wiener filter (MWF) beamformer for MI455X / gfx1250.
//
// Shapes (fixed by setup_inputs): N=8, C=16, T=300, F=257.
//   mixture : (N, 2, C, T, F) f32     target : (N, 2, C, T, F) f32
//   steering_vector: unused by the reference (ignored here too)
//   out     : (N, 2, 1, T, F) f32
//
// Bandwidth-bound (~330MB traffic -> ~14us @ 23.3 TB/s). The per-(n,f)
// 16x16 complex gram over T is computed with V_WMMA_F32_16X16X4_F32:
// 4 real WMMAs (RR, II, IR, RI) per K=4 slab; for a gram, the documented
// f32 A(16x4)/B(4x16) lane layouts coincide, so one register pair feeds
// both operands.
// ---------------------------------------------------------------------------

typedef __attribute__((ext_vector_type(2))) float v2f;
typedef __attribute__((ext_vector_type(8))) float v8f;

namespace mwf {
constexpr int N = 8, C = 16, T = 300, F = 257;
constexpr int NP = N * F;                 // 2056 (n,f) pairs
constexpr int FT = (F + 7) / 8;           // 33 f-tiles of 8 per batch entry
constexpr float LOADF = 7.0710678118654755e-04f;  // 0.001/sqrt(2)
}

__device__ __forceinline__ float2 cmul(float2 a, float2 b) {
  return make_float2(a.x * b.x - a.y * b.y, a.x * b.y + a.y * b.x);
}
__device__ __forceinline__ float2 cmulc(float2 a, float2 b) {  // a * conj(b)
  return make_float2(a.x * b.x + a.y * b.y, a.y * b.x - a.x * b.y);
}

// ---------------------------------------------------------------------------
// K1: per-(n,f) wave computes
//   phi   = Y Y^H / T + (LOAD + i LOAD) I          (16x16, via WMMA f32)
//   sab   = (1/T) sum_t y[:,t] * conj(b[t])        (16,)
//   abar  = (1/T) sum_t y[:,t]                     (16,)
//   bbar  = (1/T) sum_t b[t]                       scalar
// One wave per (n,f); 8 waves/block cover 8 consecutive f (cacheline reuse).
// ---------------------------------------------------------------------------
__global__ __launch_bounds__(256) void mwf_phi_kernel(
    const float* __restrict__ mix, const float* __restrict__ tgt,
    float2* __restrict__ phi, float2* __restrict__ sab,
    float2* __restrict__ abar, float2* __restrict__ bbar) {
  using namespace mwf;
  const int lane = threadIdx.x & 31;
  const int wave = threadIdx.x >> 5;
  const int n = blockIdx.x / FT;
  const int f = (blockIdx.x % FT) * 8 + wave;
  if (f >= F) return;  // wave-uniform: EXEC stays all-ones for WMMA

  const int m = lane & 15;      // channel (A row / B col)
  const int half = lane >> 4;   // which K pair of the 4-slab this lane holds

  const size_t TF = (size_t)T * F;
  const float* yr_b = mix + ((size_t)(n * 2 + 0) * C + m) * TF + f;
  const float* yi_b = mix + ((size_t)(n * 2 + 1) * C + m) * TF + f;
  const float* br_b = tgt + ((size_t)(n * 2 + 0) * C + 0) * TF + f;  // ref mic
  const float* bi_b = tgt + ((size_t)(n * 2 + 1) * C + 0) * TF + f;

  v8f accRR = {}, accII = {}, accIR = {}, accRI = {};
  float sr = 0.f, si = 0.f, ar_s = 0.f, ai_s = 0.f, br_s = 0.f, bi_s = 0.f;

  for (int t0 = 0; t0 < T; t0 += 4) {
    const int ta = t0 + 2 * half;  // this lane's K=2*half, 2*half+1
    float yr0 = yr_b[(size_t)ta * F];
    float yr1 = yr_b[(size_t)(ta + 1) * F];
    float yi0 = yi_b[(size_t)ta * F];
    float yi1 = yi_b[(size_t)(ta + 1) * F];
    float br0 = br_b[(size_t)ta * F];
    float br1 = br_b[(size_t)(ta + 1) * F];
    float bi0 = bi_b[(size_t)ta * F];
    float bi1 = bi_b[(size_t)(ta + 1) * F];

    v2f ar = {yr0, yr1};
    v2f ai = {yi0, yi1};
    // Gram: identical lane data is valid as both A (16x4) and B (4x16).
    accRR = __builtin_amdgcn_wmma_f32_16x16x4_f32(false, ar, false, ar,
                                                  (short)0, accRR, false, false);
    accII = __builtin_amdgcn_wmma_f32_16x16x4_f32(false, ai, false, ai,
                                                  (short)0, accII, false, false);
    accIR = __builtin_amdgcn_wmma_f32_16x16x4_f32(false, ai, false, ar,
                                                  (short)0, accIR, false, false);
    accRI = __builtin_amdgcn_wmma_f32_16x16x4_f32(false, ar, false, ai,
                                                  (short)0, accRI, false, false);

    // y * conj(b) and plain sums over this lane's two t's
    sr += yr0 * br0 + yi0 * bi0 + yr1 * br1 + yi1 * bi1;
    si += yi0 * br0 - yr0 * bi0 + yi1 * br1 - yr1 * bi1;
    ar_s += yr0 + yr1;
    ai_s += yi0 + yi1;
    br_s += br0 + br1;
    bi_s += bi0 + bi1;
  }

  // Fold the two half-wave t-partitions together (lane L <-> L^16).
  sr += __shfl_xor(sr, 16);
  si += __shfl_xor(si, 16);
  ar_s += __shfl_xor(ar_s, 16);
  ai_s += __shfl_xor(ai_s, 16);
  br_s += __shfl_xor(br_s, 16);
  bi_s += __shfl_xor(bi_s, 16);

  const float invT = 1.0f / (float)T;
  const int p = n * F + f;

  // phi C/D layout: VGPR r -> row M = r + 8*half, col N = lane%16.
  #pragma unroll
  for (int r = 0; r < 8; ++r) {
    const int row = r + 8 * half;
    float re = (accRR[r] + accII[r]) * invT;
    float im = (accIR[r] - accRI[r]) * invT;
    if (row == m) { re += LOADF; im += LOADF; }
    phi[(size_t)p * 256 + row * 16 + m] = make_float2(re, im);
  }
  if (half == 0) {
    sab[p * 16 + m] = make_float2(sr * invT, si * invT);
    abar[p * 16 + m] = make_float2(ar_s * invT, ai_s * invT);
    if (m == 0) bbar[p] = make_float2(br_s * invT, bi_s * invT);
  }
}

// ---------------------------------------------------------------------------
// K2: tiny cross-batch means  mu_a[f,c] = mean_n abar,  mu_b[f] = mean_n bbar
// ---------------------------------------------------------------------------
__global__ void mwf_mean_kernel(const float2* __restrict__ abar,
                                const float2* __restrict__ bbar,
                                float2* __restrict__ mua,
                                float2* __restrict__ mub) {
  using namespace mwf;
  const int idx = blockIdx.x * blockDim.x + threadIdx.x;
  const float invN = 1.0f / (float)N;
  if (idx < F * 16) {
    const int f = idx / 16, c = idx % 16;
    float2 s = make_float2(0.f, 0.f);
    for (int n = 0; n < N; ++n) {
      float2 v = abar[(n * F + f) * 16 + c];
      s.x += v.x; s.y += v.y;
    }
    mua[idx] = make_float2(s.x * invN, s.y * invN);
  }
  if (idx < F) {
    float2 s = make_float2(0.f, 0.f);
    for (int n = 0; n < N; ++n) {
      float2 v = bbar[n * F + idx];
      s.x += v.x; s.y += v.y;
    }
    mub[idx] = make_float2(s.x * invN, s.y * invN);
  }
}

// ---------------------------------------------------------------------------
// K3: per-(n,f) solve phi w = phi_yx0 (16x16 complex Gauss-Jordan in LDS;
// diagonal loading keeps phi well-conditioned so no pivoting needed).
// 8 waves/block, one system per wave; 2056 = 257 blocks * 8 exactly.
// Lane = (row i = lane%16, column chunk half: 0 -> j=0..8, 1 -> j=9..16).
// ---------------------------------------------------------------------------
__global__ __launch_bounds__(256) void mwf_solve_kernel(
    const float2* __restrict__ phi, const float2* __restrict__ sab,
    const float2* __restrict__ abar, const float2* __restrict__ bbar,
    const float2* __restrict__ mua, const float2* __restrict__ mub,
    float2* __restrict__ wout) {
  using namespace mwf;
  __shared__ float2 M[8][16][18];  // [wave][row][17 cols, padded to 18]
  const int lane = threadIdx.x & 31;
  const int wave = threadIdx.x >> 5;
  const int p = blockIdx.x * 8 + wave;
  const int f = p % F;
  const int i = lane & 15;
  const int halfc = lane >> 4;
  const int j0 = halfc ? 9 : 0;
  const int jn = halfc ? 8 : 9;

  #pragma unroll
  for (int j = 0; j < 8; ++j) {
    const int e = lane + j * 32;
    M[wave][e >> 4][e & 15] = phi[(size_t)p * 256 + e];
  }
  if (halfc == 0) {
    // rhs = sab - mua*conj(bbar) - abar*conj(mub) + mua*conj(mub)
    float2 s = sab[p * 16 + i];
    float2 a = abar[p * 16 + i];
    float2 b = bbar[p];
    float2 ma = mua[f * 16 + i];
    float2 mb = mub[f];
    float2 t1 = cmulc(ma, b), t2 = cmulc(a, mb), t3 = cmulc(ma, mb);
    M[wave][i][16] = make_float2(s.x - t1.x - t2.x + t3.x,
                                 s.y - t1.y - t2.y + t3.y);
  }
  __syncthreads();

  for (int k = 0; k < 16; ++k) {
    if (i == k) {  // scale pivot row by 1/pivot
      float2 pv = M[wave][k][k];
      float d = 1.0f / (pv.x * pv.x + pv.y * pv.y);
      float2 ip = make_float2(pv.x * d, -pv.y * d);
      for (int jj = 0; jj < jn; ++jj) {
        const int j = j0 + jj;
        M[wave][k][j] = cmul(M[wave][k][j], ip);
      }
    }
    __syncthreads();
    float2 fi = M[wave][i][k];
    __syncthreads();
    if (i != k) {  // eliminate
      for (int jj = 0; jj < jn; ++jj) {
        const int j = j0 + jj;
        float2 v = M[wave][i][j];
        float2 s = cmul(fi, M[wave][k][j]);
        M[wave][i][j] = make_float2(v.x - s.x, v.y - s.y);
      }
    }
    __syncthreads();
  }
  if (halfc == 0) wout[p * 16 + i] = M[wave][i][16];
}

// ---------------------------------------------------------------------------
// K4: beamform  out[n,t,f] = sum_c conj(w[n,f,c]) * y[n,c,t,f]
// Consecutive threads -> consecutive f: y loads fully coalesced; w hits L2.
// ---------------------------------------------------------------------------
__global__ __launch_bounds__(256) void mwf_bf_kernel(
    const float* __restrict__ mix, const float2* __restrict__ w,
    float* __restrict__ out) {
  using namespace mwf;
  const int idx = blockIdx.x * blockDim.x + threadIdx.x;
  if (idx >= N * T * F) return;
  const int f = idx % F;
  const int nt = idx / F;
  const int t = nt % T;
  const int n = nt / T;

  const size_t TF = (size_t)T * F;
  const float* yr = mix + (size_t)(n * 2 + 0) * C * TF + (size_t)t * F + f;
  const float* yi = mix + (size_t)(n * 2 + 1) * C * TF + (size_t)t * F + f;
  const float2* wp = w + (size_t)(n * F + f) * 16;

  float ar = 0.f, ai = 0.f;
  #pragma unroll
  for (int c = 0; c < 16; ++c) {
    float2 wc = wp[c];
    float y_r = yr[(size_t)c * TF];
    float y_i = yi[(size_t)c * TF];
    ar += wc.x * y_r + wc.y * y_i;   // Re(conj(w)*y)
    ai += wc.x * y_i - wc.y * y_r;   // Im(conj(w)*y)
  }
  out[((size_t)(n * 2 + 0) * T + t) * F + f] = ar;
  out[((size_t)(n * 2 + 1) * T + t) * F + f] = ai;
}

// ---------------------------------------------------------------------------
extern "C" void kernel_launch(void* const* d_in, const int* in_sizes, int n_in,
                              void* d_out, int out_size, void* d_ws,
                              size_t ws_size, hipStream_t stream) {
  using namespace mwf;
  (void)in_sizes; (void)n_in; (void)out_size; (void)ws_size;
  const float* mix = (const float*)d_in[0];
  const float* tgt = (const float*)d_in[1];
  // d_in[2] (steering_vector) is unused by the reference.
  float* out = (float*)d_out;

  // Workspace carve-up (~5.05 MB of float2).
  float2* phi  = (float2*)d_ws;                    // NP*256
  float2* sab  = phi + (size_t)NP * 256;           // NP*16
  float2* abar = sab + (size_t)NP * 16;            // NP*16
  float2* bbar = abar + (size_t)NP * 16;           // NP
  float2* mua  = bbar + NP;                        // F*16
  float2* mub  = mua + (size_t)F * 16;             // F
  float2* wv   = mub + F;                          // NP*16

  mwf_phi_kernel<<<N * FT, 256, 0, stream>>>(mix, tgt, phi, sab, abar, bbar);
  mwf_mean_kernel<<<(F * 16 + 255) / 256, 256, 0, stream>>>(abar, bbar, mua, mub);
  mwf_solve_kernel<<<NP / 8, 256, 0, stream>>>(phi, sab, abar, bbar, mua, mub, wv);
  mwf_bf_kernel<<<((size_t)N * T * F + 255) / 256, 256, 0, stream>>>(mix, wv, out);
}